// DistMultPredictor_6614249636085
// MI455X (gfx1250) — compile-verified
//
#include <hip/hip_runtime.h>
#include <hip/hip_bf16.h>

typedef __attribute__((ext_vector_type(2))) float v2f;
typedef __attribute__((ext_vector_type(8))) float v8f;

#define N_HID 128
#define TILE_EDGES 16

// One wave (32 lanes) computes 16 DistMult scores per tile via
// V_WMMA_F32_16X16X4_F32, accumulating D = (h_u * w) x h_v^T over K=128
// in 32 steps and extracting the diagonal.
__global__ __launch_bounds__(256)
void distmult_wmma_kernel(const float* __restrict__ h,
                          const float* __restrict__ W,
                          const int*   __restrict__ src,
                          const int*   __restrict__ dst,
                          const int*   __restrict__ rel,
                          float*       __restrict__ out,
                          int n_edges, int tiles_per_etype)
{
    __shared__ float wlds[N_HID];

    const int t = blockIdx.y;  // etype

    // Stage relation embedding row into LDS (broadcast source for all waves).
    if (threadIdx.x < N_HID) {
        const int wrow = rel[t];
        wlds[threadIdx.x] = W[wrow * N_HID + threadIdx.x];
    }
    __syncthreads();

    const int lane   = threadIdx.x & 31;
    const int wave   = threadIdx.x >> 5;
    const int wavesX = gridDim.x * (blockDim.x >> 5);
    const int eLane  = lane & 15;                 // edge-in-tile this lane feeds
    const int dimOff = (lane & 16) ? 2 : 0;       // K sub-offset for lane half

    const long long etypeBase = (long long)t * n_edges;

    for (int tile = blockIdx.x * (blockDim.x >> 5) + wave;
         tile < tiles_per_etype; tile += wavesX)
    {
        const int edge0  = tile * TILE_EDGES;
        int myEdge = edge0 + eLane;
        if (myEdge >= n_edges) myEdge = n_edges - 1;   // clamp (store is guarded)

        const long long si = src[etypeBase + myEdge];
        const long long di = dst[etypeBase + myEdge];

        const float* uRow = h + si * N_HID + dimOff;   // A source: h[src]
        const float* vRow = h + di * N_HID + dimOff;   // B source: h[dst]
        const float* wRow = wlds + dimOff;             // per-dim relation scale

        v8f c = {};  // 16x16 f32 accumulator
        #pragma unroll 8
        for (int k = 0; k < N_HID / 4; ++k) {
            // Lane L<16 supplies dims {4k,4k+1}; L>=16 supplies {4k+2,4k+3}.
            v2f u = *(const v2f*)(uRow + 4 * k);
            v2f v = *(const v2f*)(vRow + 4 * k);
            v2f w = *(const v2f*)(wRow + 4 * k);   // LDS broadcast load
            v2f a;
            a.x = u.x * w.x;
            a.y = u.y * w.y;
            // D = A(16x4) * B(4x16) + C ; full f32 precision
            c = __builtin_amdgcn_wmma_f32_16x16x4_f32(
                    false, a, false, v, (short)0, c, false, false);
        }

        // Diagonal extraction:
        //   lanes 0..7  hold D[e,e] for e = lane     in c[lane]
        //   lanes 24..31 hold D[e,e] for e = lane-16 in c[lane-24]
        const int sel = lane & 7;
        float s = c[0];
        s = (sel == 1) ? c[1] : s;
        s = (sel == 2) ? c[2] : s;
        s = (sel == 3) ? c[3] : s;
        s = (sel == 4) ? c[4] : s;
        s = (sel == 5) ? c[5] : s;
        s = (sel == 6) ? c[6] : s;
        s = (sel == 7) ? c[7] : s;

        const bool lowHalf  = (lane < 8);
        const bool highHalf = (lane >= 24);
        if (lowHalf || highHalf) {
            const int e = edge0 + (lowHalf ? lane : (lane - 16));
            if (e < n_edges) {
                out[etypeBase + e] = 1.0f / (1.0f + __expf(-s));
            }
        }
    }
}

extern "C" void kernel_launch(void* const* d_in, const int* in_sizes, int n_in,
                              void* d_out, int out_size, void* d_ws, size_t ws_size,
                              hipStream_t stream) {
    (void)n_in; (void)d_ws; (void)ws_size; (void)out_size;

    const float* h   = (const float*)d_in[0];
    const float* W   = (const float*)d_in[1];
    const int*   src = (const int*)d_in[2];
    const int*   dst = (const int*)d_in[3];
    const int*   rel = (const int*)d_in[4];
    float*       out = (float*)d_out;

    const int n_etypes = in_sizes[4];                 // rel is [T]
    const int n_edges  = in_sizes[2] / n_etypes;      // src is [T, E]
    const int tiles    = (n_edges + TILE_EDGES - 1) / TILE_EDGES;

    dim3 block(256, 1, 1);                            // 8 waves / block
    dim3 grid(98, n_etypes, 1);                       // grid-stride over tiles

    distmult_wmma_kernel<<<grid, block, 0, stream>>>(
        h, W, src, dst, rel, out, n_edges, tiles);
}